// CRF_Layer_54992761258512
// MI455X (gfx1250) — compile-verified
//
#include <hip/hip_runtime.h>
#include <stdint.h>

// Viterbi (CRF) decode, S=65536, T=6, via (max,+)-semiring parallel scan.
// CDNA5 paths used: global_load_async_to_lds_b128 / global_store_async_from_lds_b128
// (ASYNCcnt), s_wait_asynccnt / s_wait_dscnt split counters, wave32 layout,
// LDS tree-scan on a single WGP (320 KB LDS).
// NOTE: no v_wmma on purpose — the recurrence is a (max,+) semiring product;
// matrix cores implement (+,x) and cannot express max-reduction.

#define S_LEN 65536
#define TGS 6
#define NB 256          // blocks
#define CH 256          // timesteps per block (NB*CH == S_LEN)
#define START_TAG 4
#define STOP_TAG 5
#define NEGV (-10000.0f)

// workspace layout (bytes); total ~316 KB
#define WS_P 0                                   // NB*36 floats  (block (max,+) products)
#define WS_FV (WS_P + NB * 36 * 4)               // (NB+1)*8 floats (fv at block starts, stride 8)
#define WS_MAP (WS_FV + (NB + 1) * 8 * 4)        // NB*8 ints (end-tag -> start-tag maps)
#define WS_END (WS_MAP + NB * 8 * 4)             // NB ints (tag at end of each block)
#define WS_BPW (WS_END + NB * 4)                 // S_LEN uint32 (packed 4-bit backpointers)

// ---------------- CDNA5 async LDS DMA helpers ----------------

__device__ __forceinline__ uint32_t lds_off_of(const void* p) {
  // generic LDS pointer = {shared_aperture_hi32, lds_byte_offset}; low 32 bits are the LDS address
  return (uint32_t)(uintptr_t)p;
}
__device__ __forceinline__ void async_load_lds_b128(uint32_t ldsoff, uint64_t gaddr) {
  asm volatile("global_load_async_to_lds_b128 %0, %1, off"
               :: "v"(ldsoff), "v"(gaddr) : "memory");
}
__device__ __forceinline__ void async_store_lds_b128(uint64_t gaddr, uint32_t ldsoff) {
  asm volatile("global_store_async_from_lds_b128 %0, %1, off"
               :: "v"(gaddr), "v"(ldsoff) : "memory");
}
__device__ __forceinline__ void wait_async0() {
  asm volatile("s_wait_asynccnt 0" ::: "memory");
}
__device__ __forceinline__ void wait_ds0() {
  asm volatile("s_wait_dscnt 0" ::: "memory");
}

// ---------------- K1: per-block (max,+) product matrices ----------------
// Lane j (j<6) holds column j of the running product P (identity in max-plus).
// Update with M_t (M_t[i,k] = trans[i,k] + feat_t[i]) is per-lane independent
// because trans and feat_t are wave-uniform.
__global__ void k1_block_products(const float* __restrict__ feats,
                                  const float* __restrict__ trans,
                                  float* __restrict__ wsP) {
  __shared__ alignas(16) float s_feat[CH * TGS];
  const int b = blockIdx.x;
  const int lane = threadIdx.x;

  // async DMA: 6 KB feats chunk -> LDS (12 x 512B)
  uint64_t g = (uint64_t)(uintptr_t)(feats + (size_t)b * CH * TGS);
  uint32_t l = lds_off_of(s_feat);
  #pragma unroll
  for (int i = 0; i < (CH * TGS * 4) / (32 * 16); ++i)
    async_load_lds_b128(l + (uint32_t)lane * 16u + (uint32_t)i * 512u,
                        g + (uint64_t)lane * 16u + (uint64_t)i * 512u);

  float tr[TGS][TGS];  // uniform -> scalar loads, overlaps with DMA
  #pragma unroll
  for (int i = 0; i < TGS; ++i)
    #pragma unroll
    for (int k = 0; k < TGS; ++k) tr[i][k] = trans[i * TGS + k];

  wait_async0();

  const int j = lane;
  float p[TGS];
  #pragma unroll
  for (int k = 0; k < TGS; ++k) p[k] = (k == j) ? 0.0f : NEGV;

  for (int t = 0; t < CH; ++t) {
    float ft[TGS];
    #pragma unroll
    for (int i = 0; i < TGS; ++i) ft[i] = s_feat[t * TGS + i];
    float np[TGS];
    #pragma unroll
    for (int i = 0; i < TGS; ++i) {
      float m = tr[i][0] + p[0];
      #pragma unroll
      for (int k = 1; k < TGS; ++k) m = fmaxf(m, tr[i][k] + p[k]);
      np[i] = m + ft[i];
    }
    #pragma unroll
    for (int k = 0; k < TGS; ++k) p[k] = np[k];
  }
  if (j < TGS) {
    #pragma unroll
    for (int i = 0; i < TGS; ++i) wsP[b * 36 + i * TGS + j] = p[i];
  }
}

// ---------------- K2: Hillis-Steele (max,+) tree scan over block matrices ----------------
// One workgroup, 256 threads (8 waves). All NB products DMA'd into LDS once;
// 8 double-buffered composition levels; then fv at every block start is emitted.
// Replaces a 256-deep dependent-global-load chain with LDS-bandwidth work.
__global__ void k2_scan_tree(const float* __restrict__ wsP, float* __restrict__ wsFV) {
  __shared__ alignas(16) float bufA[NB * 36];
  __shared__ alignas(16) float bufB[NB * 36];
  const int b = threadIdx.x;  // 0..255, one matrix per thread

  // async DMA: 36 KB of block products -> LDS (9 x 4096B sweeps, 256 lanes x 16B)
  uint64_t g = (uint64_t)(uintptr_t)wsP;
  uint32_t l = lds_off_of(bufA);
  #pragma unroll
  for (int i = 0; i < (NB * 36 * 4) / (256 * 16); ++i)
    async_load_lds_b128(l + (uint32_t)b * 16u + (uint32_t)i * 4096u,
                        g + (uint64_t)b * 16u + (uint64_t)i * 4096u);
  wait_async0();
  __syncthreads();

  float* src = bufA;
  float* dst = bufB;
  for (int d = 1; d < NB; d <<= 1) {   // 8 levels
    float c[36];
    if (b >= d) {
      float a[36], e[36];
      const float* As = src + b * 36;        // later-in-time factor
      const float* Bs = src + (b - d) * 36;  // earlier-in-time factor
      #pragma unroll
      for (int x = 0; x < 36; ++x) a[x] = As[x];
      #pragma unroll
      for (int x = 0; x < 36; ++x) e[x] = Bs[x];
      #pragma unroll
      for (int i = 0; i < TGS; ++i)
        #pragma unroll
        for (int j = 0; j < TGS; ++j) {
          float m = a[i * TGS + 0] + e[0 * TGS + j];
          #pragma unroll
          for (int k = 1; k < TGS; ++k) m = fmaxf(m, a[i * TGS + k] + e[k * TGS + j]);
          c[i * TGS + j] = m;
        }
    } else {
      const float* As = src + b * 36;
      #pragma unroll
      for (int x = 0; x < 36; ++x) c[x] = As[x];
    }
    float* Ds = dst + b * 36;
    #pragma unroll
    for (int x = 0; x < 36; ++x) Ds[x] = c[x];
    __syncthreads();
    float* tmp = src; src = dst; dst = tmp;
  }
  // after 8 swaps, inclusive prefix Q_b = P_b (x) ... (x) P_0 lives in src (== bufA)

  // fv_start_{b+1}[n] = max_k ( Q_b[n,k] + init[k] )
  float init[TGS];
  #pragma unroll
  for (int k = 0; k < TGS; ++k) init[k] = (k == START_TAG) ? 0.0f : NEGV;

  const float* Q = src + b * 36;
  #pragma unroll
  for (int n = 0; n < TGS; ++n) {
    float m = Q[n * TGS + 0] + init[0];
    #pragma unroll
    for (int k = 1; k < TGS; ++k) m = fmaxf(m, Q[n * TGS + k] + init[k]);
    wsFV[(b + 1) * 8 + n] = m;
  }
  if (b == 0) {
    #pragma unroll
    for (int k = 0; k < TGS; ++k) wsFV[k] = init[k];  // fv_start_0
  }
}

// ---------------- K3: replay chunk, emit packed backpointers + boundary map ----------------
__global__ void k3_replay(const float* __restrict__ feats,
                          const float* __restrict__ trans,
                          const float* __restrict__ wsFV,
                          uint32_t* __restrict__ wsBPW,
                          int* __restrict__ wsMAP) {
  __shared__ alignas(16) float s_feat[CH * TGS];
  __shared__ alignas(16) uint32_t s_bpw[CH];
  const int b = blockIdx.x;
  const int lane = threadIdx.x;

  uint64_t g = (uint64_t)(uintptr_t)(feats + (size_t)b * CH * TGS);
  uint32_t l = lds_off_of(s_feat);
  #pragma unroll
  for (int i = 0; i < (CH * TGS * 4) / (32 * 16); ++i)
    async_load_lds_b128(l + (uint32_t)lane * 16u + (uint32_t)i * 512u,
                        g + (uint64_t)lane * 16u + (uint64_t)i * 512u);

  float tr[TGS][TGS];
  #pragma unroll
  for (int i = 0; i < TGS; ++i)
    #pragma unroll
    for (int k = 0; k < TGS; ++k) tr[i][k] = trans[i * TGS + k];

  float fv[TGS];  // replicated in every lane (uniform)
  #pragma unroll
  for (int k = 0; k < TGS; ++k) fv[k] = wsFV[b * 8 + k];

  wait_async0();

  for (int t = 0; t < CH; ++t) {
    float ft[TGS];
    #pragma unroll
    for (int i = 0; i < TGS; ++i) ft[i] = s_feat[t * TGS + i];
    uint32_t wrd = 0;
    float nf[TGS];
    #pragma unroll
    for (int n = 0; n < TGS; ++n) {
      float best = fv[0] + tr[n][0];
      int bi = 0;
      #pragma unroll
      for (int p = 1; p < TGS; ++p) {
        float v = fv[p] + tr[n][p];
        if (v > best) { best = v; bi = p; }   // strict '>' == jnp.argmax first-max rule
      }
      nf[n] = best + ft[n];
      wrd |= (uint32_t)bi << (4 * n);
    }
    #pragma unroll
    for (int k = 0; k < TGS; ++k) fv[k] = nf[k];
    if (lane == 0) s_bpw[t] = wrd;
  }

  wait_ds0();  // lane0's LDS writes committed before DMA engine reads LDS
  // async DMA: 1 KB backpointer words LDS -> global (2 x 512B)
  uint64_t go = (uint64_t)(uintptr_t)(wsBPW + (size_t)b * CH);
  uint32_t lo = lds_off_of(s_bpw);
  #pragma unroll
  for (int i = 0; i < 2; ++i)
    async_store_lds_b128(go + (uint64_t)lane * 16u + (uint64_t)i * 512u,
                         lo + (uint32_t)lane * 16u + (uint32_t)i * 512u);

  // local backtrack map: lane e<6 chases its chunk for end-tag e
  if (lane < TGS) {
    int tag = lane;
    for (int t = CH - 1; t >= 0; --t) tag = (int)((s_bpw[t] >> (4 * tag)) & 7u);
    wsMAP[b * 8 + lane] = tag;  // tag at time b*CH-1 given end-tag 'lane'
  }
  wait_async0();
}

// ---------------- K4: terminal argmax + cross-block tag chase (in LDS) ----------------
__global__ void k4_chase(const float* __restrict__ trans,
                         const float* __restrict__ wsFV,
                         const int* __restrict__ wsMAP,
                         int* __restrict__ wsEND,
                         float* __restrict__ out) {
  __shared__ alignas(16) int s_map[NB * 8];
  __shared__ alignas(16) int s_end[NB];
  const int lane = threadIdx.x;

  uint64_t g = (uint64_t)(uintptr_t)wsMAP;
  uint32_t l = lds_off_of(s_map);
  #pragma unroll
  for (int i = 0; i < (NB * 8 * 4) / (32 * 16); ++i)
    async_load_lds_b128(l + (uint32_t)lane * 16u + (uint32_t)i * 512u,
                        g + (uint64_t)lane * 16u + (uint64_t)i * 512u);

  // terminal = fv_final + trans[STOP,:]; first-max argmax
  float best = wsFV[NB * 8 + 0] + trans[STOP_TAG * TGS + 0];
  int bi = 0;
  #pragma unroll
  for (int p = 1; p < TGS; ++p) {
    float v = wsFV[NB * 8 + p] + trans[STOP_TAG * TGS + p];
    if (v > best) { best = v; bi = p; }
  }
  if (lane == 0) out[0] = best;  // path_score

  wait_async0();
  if (lane == 0) {
    int tag = bi;
    s_end[NB - 1] = tag;
    for (int b = NB - 1; b >= 1; --b) {
      tag = s_map[b * 8 + tag];
      s_end[b - 1] = tag;
    }
  }
  wait_ds0();
  for (int i = lane; i < NB; i += 32) wsEND[i] = s_end[i];
}

// ---------------- K5: per-block local backtrack, emit path ----------------
__global__ void k5_emit(const uint32_t* __restrict__ wsBPW,
                        const int* __restrict__ wsEND,
                        float* __restrict__ out) {
  __shared__ alignas(16) uint32_t s_bpw[CH];
  const int b = blockIdx.x;
  const int lane = threadIdx.x;

  uint64_t g = (uint64_t)(uintptr_t)(wsBPW + (size_t)b * CH);
  uint32_t l = lds_off_of(s_bpw);
  #pragma unroll
  for (int i = 0; i < 2; ++i)
    async_load_lds_b128(l + (uint32_t)lane * 16u + (uint32_t)i * 512u,
                        g + (uint64_t)lane * 16u + (uint64_t)i * 512u);
  wait_async0();

  if (lane == 0) {
    int tag = wsEND[b];  // tag at global time (b+1)*CH - 1
    for (int t = CH - 1; t >= 0; --t) {
      out[1 + b * CH + t] = (float)tag;
      tag = (int)((s_bpw[t] >> (4 * tag)) & 7u);
    }
  }
}

// ---------------- launch ----------------
extern "C" void kernel_launch(void* const* d_in, const int* in_sizes, int n_in,
                              void* d_out, int out_size, void* d_ws, size_t ws_size,
                              hipStream_t stream) {
  (void)in_sizes; (void)n_in; (void)out_size; (void)ws_size;
  const float* feats = (const float*)d_in[0];   // (1, 65536, 6) f32
  const float* trans = (const float*)d_in[1];   // (6, 6) f32, [next, prev]
  float* out = (float*)d_out;                   // [score, path(65536)]
  char* ws = (char*)d_ws;                       // needs ~316 KB

  float* P = (float*)(ws + WS_P);
  float* FV = (float*)(ws + WS_FV);
  int* MAP = (int*)(ws + WS_MAP);
  int* END = (int*)(ws + WS_END);
  uint32_t* BPW = (uint32_t*)(ws + WS_BPW);

  k1_block_products<<<NB, 32, 0, stream>>>(feats, trans, P);
  k2_scan_tree<<<1, 256, 0, stream>>>(P, FV);
  k3_replay<<<NB, 32, 0, stream>>>(feats, trans, FV, BPW, MAP);
  k4_chase<<<1, 32, 0, stream>>>(trans, FV, MAP, END, out);
  k5_emit<<<NB, 32, 0, stream>>>(BPW, END, out);
}